// KinematicLayer_17738214933166
// MI455X (gfx1250) — compile-verified
//
#include <hip/hip_runtime.h>
#include <stdint.h>

#define EPSF 1e-8f
#define HPB 128          // hands per block
#define FPH 63           // floats per hand (21 joints * 3)

typedef unsigned int u32x4 __attribute__((ext_vector_type(4)));
typedef int i32x8 __attribute__((ext_vector_type(8)));
typedef int i32x4 __attribute__((ext_vector_type(4)));

#if defined(__has_builtin)
#if __has_builtin(__builtin_amdgcn_tensor_load_to_lds) && __has_builtin(__builtin_amdgcn_s_wait_tensorcnt)
#define USE_TDM 1
#endif
#if defined(USE_TDM) && __has_builtin(__builtin_amdgcn_tensor_store_from_lds)
#define USE_TDM_STORE 1
#endif
#endif

__device__ __forceinline__ unsigned lds_byte_offset(const void* p) {
  // generic pointer to LDS: low 32 bits are the LDS byte offset (aperture in high bits)
  return (unsigned)(unsigned long long)(uintptr_t)p;
}

#ifdef USE_TDM
// Build a 1-D TDM descriptor covering nelem f32 between global gaddr and the LDS byte offset lds_off
#define TDM_DESC(ga, lds_off, nelem, g0, g1)                                        \
  u32x4 g0;                                                                         \
  g0[0] = 1u;                                  /* count=1, user mode */             \
  g0[1] = (lds_off);                           /* lds_addr (bytes) */               \
  g0[2] = (unsigned)((ga) & 0xFFFFFFFFull);    /* global_addr[31:0] */              \
  g0[3] = ((unsigned)(((ga) >> 32) & 0x01FFFFFFull)) | (2u << 30); /* +type=2 */    \
  i32x8 g1;                                                                         \
  g1[0] = (int)(2u << 16);                     /* data_size=4B */                   \
  g1[1] = (int)(((nelem) & 0xFFFFu) << 16);    /* tensor_dim0[15:0] */              \
  g1[2] = (int)(((nelem) >> 16) | (1u << 16)); /* tensor_dim0[31:16]; t_dim1=1 */   \
  g1[3] = (int)(((nelem) & 0xFFFFu) << 16);    /* tile_dim0 = nelem (<=8064) */     \
  g1[4] = 1;                                   /* tile_dim1=1 */                    \
  g1[5] = (int)(nelem);                        /* tensor_dim0_stride */             \
  g1[6] = 0;                                                                        \
  g1[7] = 0;

__device__ __forceinline__ void tdm_load_1d(const float* gsrc, unsigned lds_off, unsigned nelem) {
  unsigned long long ga = (unsigned long long)(uintptr_t)gsrc;
  TDM_DESC(ga, lds_off, nelem, g0, g1)
  i32x4 gz4 = {0, 0, 0, 0};
  i32x8 gz8 = {0, 0, 0, 0, 0, 0, 0, 0};
  __builtin_amdgcn_tensor_load_to_lds(g0, g1, gz4, gz4, gz8, 0);
}
#endif

#ifdef USE_TDM_STORE
__device__ __forceinline__ void tdm_store_1d(float* gdst, unsigned lds_off, unsigned nelem) {
  unsigned long long ga = (unsigned long long)(uintptr_t)gdst;
  TDM_DESC(ga, lds_off, nelem, g0, g1)
  i32x4 gz4 = {0, 0, 0, 0};
  i32x8 gz8 = {0, 0, 0, 0, 0, 0, 0, 0};
  __builtin_amdgcn_tensor_store_from_lds(g0, g1, gz4, gz4, gz8, 0);
}
#endif

__device__ __forceinline__ void coop_g2s(float* s, const float* g, int n, int tid, int nthr) {
  int n4 = n >> 2;
  const float4* g4 = (const float4*)g;
  float4* s4 = (float4*)s;
  for (int i = tid; i < n4; i += nthr) s4[i] = g4[i];
  for (int i = (n4 << 2) + tid; i < n; i += nthr) s[i] = g[i];
}

__device__ __forceinline__ void coop_s2g(float* g, const float* s, int n, int tid, int nthr) {
  int n4 = n >> 2;
  const float4* s4 = (const float4*)s;
  float4* g4 = (float4*)g;
  for (int i = tid; i < n4; i += nthr) g4[i] = s4[i];
  for (int i = (n4 << 2) + tid; i < n; i += nthr) g[i] = s[i];
}

// one cyclic-Jacobi rotation on symmetric 3x3 (pair p,q), accumulating V (columns p,q)
__device__ __forceinline__ void jrot(float& app, float& aqq, float& apq, float& arp, float& arq,
                                     float& vp0, float& vq0, float& vp1, float& vq1,
                                     float& vp2, float& vq2) {
  float a = apq;
  if (fabsf(a) > 1e-20f) {
    float tau = (aqq - app) / (2.0f * a);
    float t = copysignf(1.0f, tau) / (fabsf(tau) + sqrtf(1.0f + tau * tau));
    float c = 1.0f / sqrtf(1.0f + t * t);
    float s = t * c;
    app -= t * a;
    aqq += t * a;
    apq = 0.0f;
    float rp = arp, rq = arq;
    arp = c * rp - s * rq;
    arq = s * rp + c * rq;
    float x;
    x = vp0; vp0 = c * x - s * vq0; vq0 = s * x + c * vq0;
    x = vp1; vp1 = c * x - s * vq1; vq1 = s * x + c * vq1;
    x = vp2; vp2 = c * x - s * vq2; vq2 = s * x + c * vq2;
  }
}

// Rodrigues rotation taking v0 -> v1 (matches reference _rot_between incl. EPS regularization)
__device__ __forceinline__ void make_rot(float v0x, float v0y, float v0z,
                                         float v1x, float v1y, float v1z, float r[9]) {
  float i0 = 1.0f / (sqrtf(v0x * v0x + v0y * v0y + v0z * v0z) + EPSF);
  float ax = v0x * i0, ay = v0y * i0, az = v0z * i0;
  float i1 = 1.0f / (sqrtf(v1x * v1x + v1y * v1y + v1z * v1z) + EPSF);
  float bx = v1x * i1, by = v1y * i1, bz = v1z * i1;
  float c = ax * bx + ay * by + az * bz;
  float cx = ay * bz - az * by, cy = az * bx - ax * bz, cz = ax * by - ay * bx;
  float s = sqrtf(cx * cx + cy * cy + cz * cz);
  float in = 1.0f / (s + EPSF);
  float x = cx * in, y = cy * in, z = cz * in;
  float k = 1.0f - c;
  r[0] = 1.0f - k * (y * y + z * z); r[1] = -s * z + k * x * y;         r[2] =  s * y + k * x * z;
  r[3] =  s * z + k * x * y;         r[4] = 1.0f - k * (x * x + z * z); r[5] = -s * x + k * y * z;
  r[6] = -s * y + k * x * z;         r[7] =  s * x + k * y * z;         r[8] = 1.0f - k * (x * x + y * y);
}

__device__ __forceinline__ void rot_about(const float r[9], float px, float py, float pz,
                                          float& x, float& y, float& z) {
  float dx = x - px, dy = y - py, dz = z - pz;
  x = r[0] * dx + r[1] * dy + r[2] * dz + px;
  y = r[3] * dx + r[4] * dy + r[5] * dz + py;
  z = r[6] * dx + r[7] * dy + r[8] * dz + pz;
}

__global__ __launch_bounds__(HPB) void kin_match_kernel(const float* __restrict__ jgt,
                                                        const float* __restrict__ tmp,
                                                        float* __restrict__ out, int N) {
  __shared__ __align__(16) float sGT[HPB * FPH];
  __shared__ __align__(16) float sTM[HPB * FPH];
  const int tid = threadIdx.x;
  const int base = blockIdx.x * HPB;
  const int cnt = (N - base < HPB) ? (N - base) : HPB;
  const int nel = cnt * FPH;
  const size_t gofs = (size_t)base * FPH;

#ifdef USE_TDM
  if (tid < 32) {  // wave 0 drives the two TDM descriptors; EXEC is ignored by TDM
    tdm_load_1d(jgt + gofs, lds_byte_offset(sGT), (unsigned)nel);
    tdm_load_1d(tmp + gofs, lds_byte_offset(sTM), (unsigned)nel);
    __builtin_amdgcn_s_wait_tensorcnt(0);
  }
#else
  coop_g2s(sGT, jgt + gofs, nel, tid, HPB);
  coop_g2s(sTM, tmp + gofs, nel, tid, HPB);
#endif
  __syncthreads();

  if (tid < cnt) {
    const float* G = &sGT[tid * FPH];
    const float* T = &sTM[tid * FPH];
    const float wx = G[0], wy = G[1], wz = G[2];
    const float t0x = T[0], t0y = T[1], t0z = T[2];

    float tx[21], ty[21], tz[21];
    // H = tJ0^T * jg  (3x3 cross-covariance), keep tJ0 in registers
    float h00 = 0, h01 = 0, h02 = 0, h10 = 0, h11 = 0, h12 = 0, h20 = 0, h21 = 0, h22 = 0;
#pragma unroll
    for (int j = 0; j < 21; ++j) {
      float axx = T[3 * j + 0] - t0x, ayy = T[3 * j + 1] - t0y, azz = T[3 * j + 2] - t0z;
      float bxx = G[3 * j + 0] - wx, byy = G[3 * j + 1] - wy, bzz = G[3 * j + 2] - wz;
      tx[j] = axx; ty[j] = ayy; tz[j] = azz;
      h00 += axx * bxx; h01 += axx * byy; h02 += axx * bzz;
      h10 += ayy * bxx; h11 += ayy * byy; h12 += ayy * bzz;
      h20 += azz * bxx; h21 += azz * byy; h22 += azz * bzz;
    }

    // S = H^T H, Jacobi eigen -> V (right singular vectors)
    float s00 = h00 * h00 + h10 * h10 + h20 * h20;
    float s01 = h00 * h01 + h10 * h11 + h20 * h21;
    float s02 = h00 * h02 + h10 * h12 + h20 * h22;
    float s11 = h01 * h01 + h11 * h11 + h21 * h21;
    float s12 = h01 * h02 + h11 * h12 + h21 * h22;
    float s22 = h02 * h02 + h12 * h12 + h22 * h22;
    float v00 = 1, v01 = 0, v02 = 0, v10 = 0, v11 = 1, v12 = 0, v20 = 0, v21 = 0, v22 = 1;
#pragma unroll
    for (int sweep = 0; sweep < 4; ++sweep) {
      jrot(s00, s11, s01, s02, s12, v00, v01, v10, v11, v20, v21);
      jrot(s00, s22, s02, s01, s12, v00, v02, v10, v12, v20, v22);
      jrot(s11, s22, s12, s01, s02, v01, v02, v11, v12, v21, v22);
    }
    // sort eigenvalues descending (track det(V) sign through column swaps)
    float detv = 1.0f, t_;
    if (s00 < s11) { t_ = s00; s00 = s11; s11 = t_; t_ = v00; v00 = v01; v01 = t_;
                     t_ = v10; v10 = v11; v11 = t_; t_ = v20; v20 = v21; v21 = t_; detv = -detv; }
    if (s00 < s22) { t_ = s00; s00 = s22; s22 = t_; t_ = v00; v00 = v02; v02 = t_;
                     t_ = v10; v10 = v12; v12 = t_; t_ = v20; v20 = v22; v22 = t_; detv = -detv; }
    if (s11 < s22) { t_ = s11; s11 = s22; s22 = t_; t_ = v01; v01 = v02; v02 = t_;
                     t_ = v11; v11 = v12; v12 = t_; t_ = v21; v21 = v22; v22 = t_; detv = -detv; }

    // U columns: u0 = norm(H v0), u1 = GS(H v1), u2 = u0 x u1 (absorbs reflection sign)
    float w0x = h00 * v00 + h01 * v10 + h02 * v20;
    float w0y = h10 * v00 + h11 * v10 + h12 * v20;
    float w0z = h20 * v00 + h21 * v10 + h22 * v20;
    float il = 1.0f / (sqrtf(w0x * w0x + w0y * w0y + w0z * w0z) + EPSF);
    float u0x = w0x * il, u0y = w0y * il, u0z = w0z * il;
    float w1x = h00 * v01 + h01 * v11 + h02 * v21;
    float w1y = h10 * v01 + h11 * v11 + h12 * v21;
    float w1z = h20 * v01 + h21 * v11 + h22 * v21;
    float pr = u0x * w1x + u0y * w1y + u0z * w1z;
    w1x -= pr * u0x; w1y -= pr * u0y; w1z -= pr * u0z;
    il = 1.0f / (sqrtf(w1x * w1x + w1y * w1y + w1z * w1z) + EPSF);
    float u1x = w1x * il, u1y = w1y * il, u1z = w1z * il;
    float u2x = u0y * u1z - u0z * u1y;
    float u2y = u0z * u1x - u0x * u1z;
    float u2z = u0x * u1y - u0y * u1x;
    v02 *= detv; v12 *= detv; v22 *= detv;   // fold d = det(V)*det(U) correction
    // Rw = v0*u0^T + v1*u1^T + v2*u2^T
    float r00 = v00 * u0x + v01 * u1x + v02 * u2x;
    float r01 = v00 * u0y + v01 * u1y + v02 * u2y;
    float r02 = v00 * u0z + v01 * u1z + v02 * u2z;
    float r10 = v10 * u0x + v11 * u1x + v12 * u2x;
    float r11 = v10 * u0y + v11 * u1y + v12 * u2y;
    float r12 = v10 * u0z + v11 * u1z + v12 * u2z;
    float r20 = v20 * u0x + v21 * u1x + v22 * u2x;
    float r21 = v20 * u0y + v21 * u1y + v22 * u2y;
    float r22 = v20 * u0z + v21 * u1z + v22 * u2z;

#pragma unroll
    for (int j = 1; j < 21; ++j) {   // tJ = Rw * tJ0 (joint 0 stays at origin)
      float xx = tx[j], yy = ty[j], zz = tz[j];
      tx[j] = r00 * xx + r01 * yy + r02 * zz;
      ty[j] = r10 * xx + r11 * yy + r12 * zz;
      tz[j] = r20 * xx + r21 * yy + r22 * zz;
    }

    // 5 independent finger chains; FK pass is algebraically identical to final tJ
#pragma unroll
    for (int f = 0; f < 5; ++f) {
      const int A = 3 * f + 1, B = 3 * f + 2, C = 3 * f + 3;
      const int TP = (f == 0) ? 17 : (f == 1) ? 18 : (f == 2) ? 20 : (f == 3) ? 19 : 16;
      float r[9];
      // step 1: parent A aligns B; rotate {B, C, TP} about tJ[A]
      make_rot(tx[B] - tx[A], ty[B] - ty[A], tz[B] - tz[A],
               (G[3 * B + 0] - wx) - tx[A], (G[3 * B + 1] - wy) - ty[A], (G[3 * B + 2] - wz) - tz[A], r);
      rot_about(r, tx[A], ty[A], tz[A], tx[B], ty[B], tz[B]);
      rot_about(r, tx[A], ty[A], tz[A], tx[C], ty[C], tz[C]);
      rot_about(r, tx[A], ty[A], tz[A], tx[TP], ty[TP], tz[TP]);
      // step 2: parent B aligns C; rotate {C, TP} about tJ[B]
      make_rot(tx[C] - tx[B], ty[C] - ty[B], tz[C] - tz[B],
               (G[3 * C + 0] - wx) - tx[B], (G[3 * C + 1] - wy) - ty[B], (G[3 * C + 2] - wz) - tz[B], r);
      rot_about(r, tx[B], ty[B], tz[B], tx[C], ty[C], tz[C]);
      rot_about(r, tx[B], ty[B], tz[B], tx[TP], ty[TP], tz[TP]);
      // step 3: parent C aligns TP; rotate {TP} about tJ[C]
      make_rot(tx[TP] - tx[C], ty[TP] - ty[C], tz[TP] - tz[C],
               (G[3 * TP + 0] - wx) - tx[C], (G[3 * TP + 1] - wy) - ty[C], (G[3 * TP + 2] - wz) - tz[C], r);
      rot_about(r, tx[C], ty[C], tz[C], tx[TP], ty[TP], tz[TP]);
    }

    // stage result (pos = tJ + wrist) into LDS for coalesced store
    float* O = &sTM[tid * FPH];
#pragma unroll
    for (int j = 0; j < 21; ++j) {
      O[3 * j + 0] = tx[j] + wx;
      O[3 * j + 1] = ty[j] + wy;
      O[3 * j + 2] = tz[j] + wz;
    }
  }
  __syncthreads();   // all ds_store results visible before the store DMA / coop store

#ifdef USE_TDM_STORE
  if (tid < 32) {    // wave 0 drives the store DMA; implicit wait-idle at s_endpgm also covers it
    tdm_store_1d(out + gofs, lds_byte_offset(sTM), (unsigned)nel);
    __builtin_amdgcn_s_wait_tensorcnt(0);
  }
#else
  coop_s2g(out + gofs, sTM, nel, tid, HPB);
#endif
}

extern "C" void kernel_launch(void* const* d_in, const int* in_sizes, int n_in,
                              void* d_out, int out_size, void* d_ws, size_t ws_size,
                              hipStream_t stream) {
  const float* jgt = (const float*)d_in[0];
  const float* tmp = (const float*)d_in[1];
  float* out = (float*)d_out;
  const int N = in_sizes[0] / FPH;  // 400000
  const int blocks = (N + HPB - 1) / HPB;
  kin_match_kernel<<<blocks, HPB, 0, stream>>>(jgt, tmp, out, N);
}